// PlasticityNetwork_48704929137328
// MI455X (gfx1250) — compile-verified
//
#include <hip/hip_runtime.h>
#include <hip/hip_bf16.h>

// CDNA5 / gfx1250 WMMA types
typedef __attribute__((ext_vector_type(2))) float v2f;   // A/B of 16x16x4 f32: 64 elems / 32 lanes
typedef __attribute__((ext_vector_type(8))) float v8f;   // C/D 16x16 f32: 256 elems / 32 lanes

// Problem dims (fixed by reference): B=64, IN=512, H=512, HID=16
#define BATCH 64
#define IN_DIM 512
#define H_DIM 512

__device__ __forceinline__ float relu_asm(float x) {
    // Single v_max_num_f32 dst, src, 0.  WMMA outputs are already canonical
    // (RNE, NaN-propagating, denorm-preserving per ISA 7.12), so the
    // compiler-inserted canonicalize max is dead weight; inline asm elides it.
    float y;
    asm("v_max_num_f32 %0, %1, 0" : "=v"(y) : "v"(x));
    return y;
}

// One wave32 per PAIR of output tiles: (i, j0..j0+15) and (i, j0+16..j0+31).
//   WMMA row m  <-> output column j0+m (for fixed i, fixed b)
//   WMMA col n  <-> hidden unit n (0..15)
// A (16x4 f32) lane layout: lanes 0-15 hold K=0,1 ; lanes 16-31 hold K=2,3
//   K0 = pre[b,i] (uniform), K1 = post[b, j0+m], K2 = weight[i, j0+m], K3 = 1.0 (bias)
// B (4x16 f32) lane layout: VGPR v, lanes 0-15 -> row K=v, lanes 16-31 -> row K=v+2
//   B = [ W1[:,0] ; W1[:,1] ; W1[:,2] ; b1 ]  (shared by both tiles, batch-invariant)
__global__ __launch_bounds__(256)
void plasticity_wmma_kernel(const float* __restrict__ pre,
                            const float* __restrict__ post,
                            const float* __restrict__ weight,
                            const float* __restrict__ W1,   // (16,3)
                            const float* __restrict__ b1,   // (16,)
                            const float* __restrict__ W2,   // (1,16)
                            const float* __restrict__ b2,   // (1,)
                            float* __restrict__ out)        // (512,512)
{
    const int lane = threadIdx.x & 31;
    const int wave = threadIdx.x >> 5;                 // 8 waves / block
    const int pair = blockIdx.x * 8 + wave;            // 512*16 = 8192 tile-pairs
    const int i    = pair >> 4;                        // 0..511
    const int j0   = (pair & 15) << 5;                 // 0..480 step 32 (covers j0..j0+31)

    const int  n  = lane & 15;                         // hidden index (column N)
    const bool hi = lane >= 16;                        // upper half-wave: K=2,3 of A / B

    // ---- batch-invariant operands (built once) ----
    v2f bmat;
    bmat.x = hi ? W1[n * 3 + 2] : W1[n * 3 + 0];       // rows K=2 / K=0
    bmat.y = hi ? b1[n]         : W1[n * 3 + 1];       // rows K=3 / K=1

    const float w2  = W2[n];                           // layer-2 weight for column n
    const float wv0 = weight[i * H_DIM + j0 + n];      // tile 0: weight[i, j0+n]
    const float wv1 = weight[i * H_DIM + j0 + 16 + n]; // tile 1: weight[i, j0+16+n]

    v8f acc0 = {0.f, 0.f, 0.f, 0.f, 0.f, 0.f, 0.f, 0.f};
    v8f acc1 = {0.f, 0.f, 0.f, 0.f, 0.f, 0.f, 0.f, 0.f};
    const v8f czero = {0.f, 0.f, 0.f, 0.f, 0.f, 0.f, 0.f, 0.f};

    #pragma unroll 2
    for (int b = 0; b < BATCH; ++b) {
        const float prev   = pre[b * IN_DIM + i];                // wave-uniform
        const float postv0 = post[b * H_DIM + j0 + n];           // per-lane, tile 0
        const float postv1 = post[b * H_DIM + j0 + 16 + n];      // per-lane, tile 1

        if (b + 1 < BATCH) {
            // gfx1250 global_prefetch_b8 for the next batch's post row
            __builtin_prefetch(&post[(b + 1) * H_DIM + j0 + n], 0, 0);
        }

        v2f a0, a1;
        a0.x = hi ? wv0  : prev;                                 // K2 : K0
        a0.y = hi ? 1.0f : postv0;                               // K3 : K1
        a1.x = hi ? wv1  : prev;
        a1.y = hi ? 1.0f : postv1;

        // D = A(16x4) x B(4x16):  D[m][n] = pre*W1[n,0]+post*W1[n,1]+w*W1[n,2]+b1[n]
        v8f d0 = __builtin_amdgcn_wmma_f32_16x16x4_f32(
            false, a0, false, bmat, (short)0, czero, false, false);
        v8f d1 = __builtin_amdgcn_wmma_f32_16x16x4_f32(
            false, a1, false, bmat, (short)0, czero, false, false);

        // layer 2 partial: acc += relu(D) * W2[n]   (reduction over n deferred)
        #pragma unroll
        for (int v = 0; v < 8; ++v) {
            acc0[v] += relu_asm(d0[v]) * w2;
            acc1[v] += relu_asm(d1[v]) * w2;
        }
    }

    // ---- reduce over hidden dim (16 lanes within each half-wave) ----
    const float bias2 = b2[0];
    #pragma unroll
    for (int v = 0; v < 8; ++v) {
        float r0 = acc0[v];
        float r1 = acc1[v];
        r0 += __shfl_xor(r0, 1, 32);  r1 += __shfl_xor(r1, 1, 32);
        r0 += __shfl_xor(r0, 2, 32);  r1 += __shfl_xor(r1, 2, 32);
        r0 += __shfl_xor(r0, 4, 32);  r1 += __shfl_xor(r1, 4, 32);
        r0 += __shfl_xor(r0, 8, 32);  r1 += __shfl_xor(r1, 8, 32);
        // lanes 0-15 now all hold row M=v; lanes 16-31 hold row M=v+8
        if (n == 0) {
            const int m = v + (hi ? 8 : 0);
            out[i * H_DIM + j0 + m]      = r0 * (1.0f / (float)BATCH) + bias2;
            out[i * H_DIM + j0 + 16 + m] = r1 * (1.0f / (float)BATCH) + bias2;
        }
    }
}

extern "C" void kernel_launch(void* const* d_in, const int* in_sizes, int n_in,
                              void* d_out, int out_size, void* d_ws, size_t ws_size,
                              hipStream_t stream) {
    const float* pre    = (const float*)d_in[0];  // (64,512)
    const float* post   = (const float*)d_in[1];  // (64,512)
    const float* weight = (const float*)d_in[2];  // (512,512)
    const float* W1     = (const float*)d_in[3];  // (16,3)
    const float* b1     = (const float*)d_in[4];  // (16,)
    const float* W2     = (const float*)d_in[5];  // (1,16)
    const float* b2     = (const float*)d_in[6];  // (1,)
    float* out          = (float*)d_out;          // (512,512)

    (void)in_sizes; (void)n_in; (void)out_size; (void)d_ws; (void)ws_size;

    // 8192 tile-pairs (512 i x 16 j-pair-blocks), 8 wave32 pairs per 256-thread block
    dim3 grid(8192 / 8);
    dim3 block(256);
    plasticity_wmma_kernel<<<grid, block, 0, stream>>>(pre, post, weight, W1, b1, W2, b2, out);
}